// mysoftmax_18476949308045
// MI455X (gfx1250) — compile-verified
//
#include <hip/hip_runtime.h>

typedef __attribute__((ext_vector_type(2))) float v2f;
typedef __attribute__((ext_vector_type(4))) float v4f;
typedef __attribute__((ext_vector_type(8))) float v8f;

#define C_DIM 128
#define S_PAD 132  // padded LDS row stride (floats): 16B-aligned rows, bank-conflict-free strided access

// Sigma_out[i,l] = p_i * p_l * ( Sigma[i,l] - u_i - v_l + s )
// with p = softmax(mu), u = Sigma p, v = Sigma^T p, s = p^T Sigma p.
// One block per batch row; 256 threads = 8 wave32.
__global__ __launch_bounds__(256) void softmax_cov_sandwich_kernel(
    const float* __restrict__ mu_in,
    const float* __restrict__ Sigma_in,
    float* __restrict__ mu_out,
    float* __restrict__ Sigma_out)
{
    extern __shared__ float lds[];
    float* sig = lds;                        // 128 * 132 floats (padded Sigma tile)
    float* p   = lds + C_DIM * S_PAD;        // 128
    float* u   = p + C_DIM;                  // 128
    float* v   = u + C_DIM;                  // 128
    float* red = v + C_DIM;                  // 128 (reduction scratch)

    const int t = threadIdx.x;
    const int b = blockIdx.x;
    const float* sigIn  = Sigma_in  + (size_t)b * C_DIM * C_DIM;
    float*       sigOut = Sigma_out + (size_t)b * C_DIM * C_DIM;

    // ---------------- softmax over the 128-class row ----------------
    float x = 0.0f;
    if (t < C_DIM) { x = mu_in[(size_t)b * C_DIM + t]; red[t] = x; }
    __syncthreads();
    for (int off = 64; off > 0; off >>= 1) {
        if (t < off) red[t] = fmaxf(red[t], red[t + off]);
        __syncthreads();
    }
    const float mx = red[0];
    __syncthreads();
    float e = 0.0f;
    if (t < C_DIM) { e = __expf(x - mx); red[t] = e; }
    __syncthreads();
    for (int off = 64; off > 0; off >>= 1) {
        if (t < off) red[t] += red[t + off];
        __syncthreads();
    }
    const float inv = 1.0f / red[0];
    __syncthreads();
    if (t < C_DIM) {
        const float pv = e * inv;
        p[t] = pv;
        mu_out[(size_t)b * C_DIM + t] = pv;
    }

    // ------- stage the 64KB Sigma tile into LDS (streamed, non-temporal) -------
    #pragma unroll 4
    for (int it = 0; it < 16; ++it) {
        const int idx = it * 256 + t;        // float4 index 0..4095 (row-major)
        const int r  = idx >> 5;
        const int c4 = idx & 31;
        const v4f g = __builtin_nontemporal_load((const v4f*)sigIn + idx);
        *(v4f*)(sig + r * S_PAD + c4 * 4) = g;
    }
    __syncthreads();

    // ---------------- WMMA: u = Sigma p  and  v = Sigma^T p ----------------
    // Wave w owns row-block (for u) and col-block (for v) [16w, 16w+16).
    // V_WMMA_F32_16X16X4_F32 fragment layout (wave32):
    //   A 16x4: lanes 0-15 hold M=lane, K={0,1} in vgpr{0,1}; lanes 16-31 hold K={2,3}
    //   B 4x16: mirrored (lane half selects K pair, lane&15 = N)
    //   D 16x16: vgpr r -> M=r (lanes 0-15) / M=8+r (lanes 16-31), N = lane&15
    const int wave = t >> 5;
    const int lane = t & 31;
    const int kh   = lane >> 4;    // which K-pair this lane supplies
    const int n16  = lane & 15;    // M index (A) / N index (B)
    const int i0   = wave * 16;

    v8f accU = {};                 // every column n equals u[i0 + m]
    v8f accV = {};                 // every row m equals v[i0 + n]
    for (int k0 = 0; k0 < C_DIM; k0 += 4) {
        // broadcast fragment: value depends only on K -> p[k0 + 2*kh + {0,1}]
        v2f pk;
        pk.x = p[k0 + 2 * kh];
        pk.y = p[k0 + 2 * kh + 1];

        // A = Sigma[i0+m, k0+2*kh .. +1]   (ds_load_b64, conflict-free via S_PAD)
        const v2f a = *(const v2f*)(sig + (i0 + n16) * S_PAD + k0 + 2 * kh);
        accU = __builtin_amdgcn_wmma_f32_16x16x4_f32(false, a, false, pk,
                                                     (short)0, accU, false, false);

        // B = Sigma[k0+2*kh .. +1, i0+n]
        v2f bc;
        bc.x = sig[(k0 + 2 * kh)     * S_PAD + i0 + n16];
        bc.y = sig[(k0 + 2 * kh + 1) * S_PAD + i0 + n16];
        accV = __builtin_amdgcn_wmma_f32_16x16x4_f32(false, pk, false, bc,
                                                     (short)0, accV, false, false);
    }
    // extract u: lane 0 holds M=0..7 in accU[0..7]; lane 16 holds M=8..15
    if (n16 == 0) {
        #pragma unroll
        for (int r = 0; r < 8; ++r) u[i0 + kh * 8 + r] = accU[r];
    }
    // extract v: row M=0 lives in accV[0] of lanes 0..15
    if (kh == 0) v[i0 + n16] = accV[0];
    __syncthreads();

    // ---------------- s = p^T u ----------------
    if (t < C_DIM) red[t] = p[t] * u[t];
    __syncthreads();
    for (int off = 64; off > 0; off >>= 1) {
        if (t < off) red[t] += red[t + off];
        __syncthreads();
    }
    const float s = red[0];

    // -------- fused output: Sigma_out = p_i p_l (Sigma - u_i - v_l + s) --------
    #pragma unroll 4
    for (int it = 0; it < 16; ++it) {
        const int idx = it * 256 + t;
        const int r = idx >> 5;
        const int c = (idx & 31) * 4;
        const v4f g = *(const v4f*)(sig + r * S_PAD + c);
        const float pr = p[r];
        const float ur = u[r];
        v4f o;
        o.x = pr * p[c + 0] * (g.x - ur - v[c + 0] + s);
        o.y = pr * p[c + 1] * (g.y - ur - v[c + 1] + s);
        o.z = pr * p[c + 2] * (g.z - ur - v[c + 2] + s);
        o.w = pr * p[c + 3] * (g.w - ur - v[c + 3] + s);
        __builtin_nontemporal_store(o, (v4f*)sigOut + idx);
    }
}

extern "C" void kernel_launch(void* const* d_in, const int* in_sizes, int n_in,
                              void* d_out, int out_size, void* d_ws, size_t ws_size,
                              hipStream_t stream) {
    (void)in_sizes; (void)n_in; (void)out_size; (void)d_ws; (void)ws_size;
    const int B = 8192;
    const float* mu_in    = (const float*)d_in[0];
    const float* Sigma_in = (const float*)d_in[1];
    float* mu_out    = (float*)d_out;                       // [B, 128]
    float* Sigma_out = (float*)d_out + (size_t)B * C_DIM;   // [B, 128, 128]

    const size_t shmem = (size_t)(C_DIM * S_PAD + 4 * C_DIM) * sizeof(float); // ~68 KB
    softmax_cov_sandwich_kernel<<<dim3(B), dim3(256), shmem, stream>>>(
        mu_in, Sigma_in, mu_out, Sigma_out);
}